// PredYWithS_55508157333621
// MI455X (gfx1250) — compile-verified
//
#include <hip/hip_runtime.h>

// ---------------------------------------------------------------------------
// Shapes: N=2048, A=256, C=100, LAT=128.  All fp32.
// Algorithm (exact rewrite of reference, 13x fewer FLOPs):
//   G[c]   = Wfc @ cov[c]                    (100 GEMMs, M=112pad,N=256,K=256)
//   D[c,i] = G[c][i] . Wfc[i]
//   E[c,i] = G[c][i] . Wfc[c]
//   H      = s @ W1^T + b1                   (M=2048,N=128,K=256)
//   BN     : scale = gamma*rsqrt(var+eps), shift = beta - mu*scale
//   F      = relu( relu_bn(H) @ W2^T + b2 )  (M=2048,N=256,K=128)
//   Z      = F @ Wfc^T + bfc                 (M=2048,N=112pad,K=256)
//   out[n,i] = Z[n,i] + ratio*0.5*(D[y,i] - 2 E[y,i] + E[y,y])
// GEMMs use V_WMMA_F32_16X16X4_F32, wave-per-16x16-tile, branch-free K loops
// (out-of-range rows clamped to valid addresses; padded outputs never read),
// dual accumulators to break the WMMA RAW chain.
// ---------------------------------------------------------------------------

typedef float v2f __attribute__((ext_vector_type(2)));
typedef float v8f __attribute__((ext_vector_type(8)));

#define WMMA_F32(a, b, c) \
  __builtin_amdgcn_wmma_f32_16x16x4_f32(false, (a), false, (b), (short)0, (c), false, false)

__device__ __forceinline__ v8f splat8(float x) {
  v8f v;
#pragma unroll
  for (int i = 0; i < 8; ++i) v[i] = x;
  return v;
}

// ---------------------------------------------------------------------------
// K1: G[c] = Wfc @ cov[c].  A = Wfc [100x256] (rows clamped), B = cov[c]
// [256x256] row-major (b32 loads coalesced across the 16 N-lanes).
// One wave per 16x16 tile of G[c] ([112 x 256] padded).
// ---------------------------------------------------------------------------
__global__ __launch_bounds__(256) void kG(const float* __restrict__ Wfc,
                                          const float* __restrict__ cov,
                                          float* __restrict__ G) {
  const int wave = (blockIdx.x * blockDim.x + threadIdx.x) >> 5;  // 0..11199
  const int lane = threadIdx.x & 31;
  const int h = lane >> 4, r = lane & 15;
  const int nt = wave % 16;          // a-tile (cols of G)
  const int mt = (wave / 16) % 7;    // i-tile (rows of G)
  const int c  = wave / (16 * 7);    // class
  const int m = min(mt * 16 + r, 99);  // clamped row of Wfc (pad rows -> garbage, unread)
  const int n = nt * 16 + r;           // col (a)
  const float* __restrict__ A = Wfc + (size_t)m * 256;
  const float* __restrict__ B = cov + (size_t)c * 256 * 256;
  v8f acc0 = splat8(0.f), acc1 = splat8(0.f);
#pragma unroll 4
  for (int k0 = 0; k0 < 256; k0 += 8) {
    const int k = k0 + 2 * h;
    v2f a0 = *(const v2f*)(A + k);
    v2f a1 = *(const v2f*)(A + k + 4);
    v2f b0, b1;
    b0.x = B[k * 256 + n];
    b0.y = B[(k + 1) * 256 + n];
    b1.x = B[(k + 4) * 256 + n];
    b1.y = B[(k + 5) * 256 + n];
    acc0 = WMMA_F32(a0, b0, acc0);
    acc1 = WMMA_F32(a1, b1, acc1);
  }
  const v8f acc = acc0 + acc1;
  float* __restrict__ Gc = G + (size_t)c * 112 * 256;
#pragma unroll
  for (int v = 0; v < 8; ++v)
    Gc[(mt * 16 + v + 8 * h) * 256 + n] = acc[v];
}

// ---------------------------------------------------------------------------
// K2: H = s @ W1^T + b1.  M=2048 (128 tiles), N=128 (8 tiles), K=256.
// ---------------------------------------------------------------------------
__global__ __launch_bounds__(256) void kH(const float* __restrict__ s,
                                          const float* __restrict__ W1,
                                          const float* __restrict__ b1,
                                          float* __restrict__ H) {
  const int wave = (blockIdx.x * blockDim.x + threadIdx.x) >> 5;  // 0..1023
  const int lane = threadIdx.x & 31;
  const int h = lane >> 4, r = lane & 15;
  const int nt = wave % 8;
  const int mt = wave / 8;
  const float* __restrict__ A = s + (size_t)(mt * 16 + r) * 256;
  const float* __restrict__ B = W1 + (size_t)(nt * 16 + r) * 256;
  v8f acc0 = splat8(b1[nt * 16 + r]), acc1 = splat8(0.f);
#pragma unroll 4
  for (int k0 = 0; k0 < 256; k0 += 8) {
    const int k = k0 + 2 * h;
    v2f a0 = *(const v2f*)(A + k);
    v2f a1 = *(const v2f*)(A + k + 4);
    v2f b0 = *(const v2f*)(B + k);
    v2f b1 = *(const v2f*)(B + k + 4);
    acc0 = WMMA_F32(a0, b0, acc0);
    acc1 = WMMA_F32(a1, b1, acc1);
  }
  const v8f acc = acc0 + acc1;
  const int n = nt * 16 + r;
#pragma unroll
  for (int v = 0; v < 8; ++v)
    H[(size_t)(mt * 16 + v + 8 * h) * 128 + n] = acc[v];
}

// ---------------------------------------------------------------------------
// K3: BatchNorm batch stats (biased var) folded into scale/shift.
// One block per feature column j in [0,128).
// ---------------------------------------------------------------------------
__global__ __launch_bounds__(256) void kBN(const float* __restrict__ H,
                                           const float* __restrict__ gamma,
                                           const float* __restrict__ beta,
                                           float* __restrict__ scale,
                                           float* __restrict__ shift) {
  const int j = blockIdx.x;
  float s1 = 0.f, s2 = 0.f;
  for (int row = threadIdx.x; row < 2048; row += 256) {
    float v = H[(size_t)row * 128 + j];
    s1 += v;
    s2 += v * v;
  }
  __shared__ float a1[256], a2[256];
  a1[threadIdx.x] = s1;
  a2[threadIdx.x] = s2;
  __syncthreads();
  for (int st = 128; st > 0; st >>= 1) {
    if (threadIdx.x < st) {
      a1[threadIdx.x] += a1[threadIdx.x + st];
      a2[threadIdx.x] += a2[threadIdx.x + st];
    }
    __syncthreads();
  }
  if (threadIdx.x == 0) {
    const float mu = a1[0] * (1.f / 2048.f);
    const float var = a2[0] * (1.f / 2048.f) - mu * mu;
    const float sc = gamma[j] * rsqrtf(var + 1e-5f);
    scale[j] = sc;
    shift[j] = beta[j] - mu * sc;
  }
}

// ---------------------------------------------------------------------------
// K4: F = relu( relu(H*scale+shift) @ W2^T + b2 ).  M=2048, N=256, K=128.
// BN+ReLU applied to the A operand in-flight.
// ---------------------------------------------------------------------------
__global__ __launch_bounds__(256) void kF(const float* __restrict__ H,
                                          const float* __restrict__ scale,
                                          const float* __restrict__ shift,
                                          const float* __restrict__ W2,
                                          const float* __restrict__ b2,
                                          float* __restrict__ F) {
  const int wave = (blockIdx.x * blockDim.x + threadIdx.x) >> 5;  // 0..2047
  const int lane = threadIdx.x & 31;
  const int h = lane >> 4, r = lane & 15;
  const int nt = wave % 16;
  const int mt = wave / 16;
  const float* __restrict__ A = H + (size_t)(mt * 16 + r) * 128;
  const float* __restrict__ B = W2 + (size_t)(nt * 16 + r) * 128;
  v8f acc0 = splat8(b2[nt * 16 + r]), acc1 = splat8(0.f);
#pragma unroll 4
  for (int k0 = 0; k0 < 128; k0 += 8) {
    const int k = k0 + 2 * h;
    v2f ar0 = *(const v2f*)(A + k);
    v2f ar1 = *(const v2f*)(A + k + 4);
    v2f a0, a1;
    a0.x = fmaxf(ar0.x * scale[k] + shift[k], 0.f);
    a0.y = fmaxf(ar0.y * scale[k + 1] + shift[k + 1], 0.f);
    a1.x = fmaxf(ar1.x * scale[k + 4] + shift[k + 4], 0.f);
    a1.y = fmaxf(ar1.y * scale[k + 5] + shift[k + 5], 0.f);
    v2f b0 = *(const v2f*)(B + k);
    v2f b1 = *(const v2f*)(B + k + 4);
    acc0 = WMMA_F32(a0, b0, acc0);
    acc1 = WMMA_F32(a1, b1, acc1);
  }
  const v8f acc = acc0 + acc1;
  const int n = nt * 16 + r;
#pragma unroll
  for (int v = 0; v < 8; ++v)
    F[(size_t)(mt * 16 + v + 8 * h) * 256 + n] = fmaxf(acc[v], 0.f);
}

// ---------------------------------------------------------------------------
// K5: Z = F @ Wfc^T + bfc.  M=2048, N=112 (7 tiles, rows of Wfc clamped;
// padded Z columns hold garbage and are never read).  K=256.
// ---------------------------------------------------------------------------
__global__ __launch_bounds__(256) void kZ(const float* __restrict__ F,
                                          const float* __restrict__ Wfc,
                                          const float* __restrict__ bfc,
                                          float* __restrict__ Z) {
  const int wave = (blockIdx.x * blockDim.x + threadIdx.x) >> 5;  // 0..895
  const int lane = threadIdx.x & 31;
  const int h = lane >> 4, r = lane & 15;
  const int nt = wave % 7;
  const int mt = wave / 7;
  const int n = min(nt * 16 + r, 99);  // clamped class row
  const float* __restrict__ A = F + (size_t)(mt * 16 + r) * 256;
  const float* __restrict__ B = Wfc + (size_t)n * 256;
  v8f acc0 = splat8(bfc[n]), acc1 = splat8(0.f);
#pragma unroll 4
  for (int k0 = 0; k0 < 256; k0 += 8) {
    const int k = k0 + 2 * h;
    v2f a0 = *(const v2f*)(A + k);
    v2f a1 = *(const v2f*)(A + k + 4);
    v2f b0 = *(const v2f*)(B + k);
    v2f b1 = *(const v2f*)(B + k + 4);
    acc0 = WMMA_F32(a0, b0, acc0);
    acc1 = WMMA_F32(a1, b1, acc1);
  }
  const v8f acc = acc0 + acc1;
  const int nc = nt * 16 + r;
#pragma unroll
  for (int v = 0; v < 8; ++v)
    Z[(size_t)(mt * 16 + v + 8 * h) * 112 + nc] = acc[v];
}

// ---------------------------------------------------------------------------
// K6: D[c,i] = G[c][i].Wfc[i] ; E[c,i] = G[c][i].Wfc[c].
// One wave per (c,i): coalesced 128B reads + shfl_xor butterfly reduction.
// ---------------------------------------------------------------------------
__global__ __launch_bounds__(256) void kDE(const float* __restrict__ G,
                                           const float* __restrict__ Wfc,
                                           float* __restrict__ D,
                                           float* __restrict__ E) {
  const int wave = (blockIdx.x * blockDim.x + threadIdx.x) >> 5;  // 0..9999
  const int lane = threadIdx.x & 31;
  const int c = wave / 100, i = wave % 100;
  const float* __restrict__ g = G + ((size_t)c * 112 + i) * 256;
  const float* __restrict__ wi = Wfc + (size_t)i * 256;
  const float* __restrict__ wc = Wfc + (size_t)c * 256;
  float d = 0.f, e = 0.f;
#pragma unroll
  for (int t = 0; t < 8; ++t) {
    const int a = lane + 32 * t;
    const float gv = g[a];
    d += gv * wi[a];
    e += gv * wc[a];
  }
#pragma unroll
  for (int off = 16; off > 0; off >>= 1) {
    d += __shfl_xor(d, off, 32);
    e += __shfl_xor(e, off, 32);
  }
  if (lane == 0) {
    D[wave] = d;
    E[wave] = e;
  }
}

// ---------------------------------------------------------------------------
// K7: out[n,i] = Z[n,i] + ratio*0.5*(D[y,i] - 2 E[y,i] + E[y,y])
// ---------------------------------------------------------------------------
__global__ __launch_bounds__(256) void kOut(const float* __restrict__ Z,
                                            const int* __restrict__ y,
                                            const float* __restrict__ ratio,
                                            const float* __restrict__ D,
                                            const float* __restrict__ E,
                                            float* __restrict__ out) {
  const int idx = blockIdx.x * blockDim.x + threadIdx.x;
  if (idx >= 2048 * 100) return;
  const int n = idx / 100, i = idx % 100;
  const int c = y[n];
  const float sg = D[c * 100 + i] - 2.f * E[c * 100 + i] + E[c * 100 + c];
  out[idx] = Z[(size_t)n * 112 + i] + ratio[0] * 0.5f * sg;
}

// ---------------------------------------------------------------------------
extern "C" void kernel_launch(void* const* d_in, const int* in_sizes, int n_in,
                              void* d_out, int out_size, void* d_ws, size_t ws_size,
                              hipStream_t stream) {
  (void)in_sizes; (void)n_in; (void)out_size; (void)ws_size;
  const float* s     = (const float*)d_in[0];
  const int*   y     = (const int*)d_in[1];
  const float* ratio = (const float*)d_in[2];
  const float* W1    = (const float*)d_in[3];
  const float* b1    = (const float*)d_in[4];
  const float* gamma = (const float*)d_in[5];
  const float* beta  = (const float*)d_in[6];
  const float* W2    = (const float*)d_in[7];
  const float* b2    = (const float*)d_in[8];
  const float* Wfc   = (const float*)d_in[9];
  const float* bfc   = (const float*)d_in[10];
  const float* cov   = (const float*)d_in[11];
  float* out = (float*)d_out;

  // Workspace layout (floats), total ~15.6 MB
  float* ws = (float*)d_ws;
  float* G     = ws;                       // 100*112*256 = 2,867,200
  float* H     = G + 2867200;              // 2048*128    =   262,144
  float* Fbuf  = H + 262144;               // 2048*256    =   524,288
  float* Z     = Fbuf + 524288;            // 2048*112    =   229,376
  float* scale = Z + 229376;               // 128
  float* shift = scale + 128;              // 128
  float* D     = shift + 128;              // 10,000
  float* E     = D + 10000;                // 10,000

  // Class-table precompute (independent of MLP path)
  kG<<<1400, 256, 0, stream>>>(Wfc, cov, G);    // 11200 waves
  kDE<<<1250, 256, 0, stream>>>(G, Wfc, D, E);  // 10000 waves

  // MLP forward
  kH<<<128, 256, 0, stream>>>(s, W1, b1, H);    // 1024 waves
  kBN<<<128, 256, 0, stream>>>(H, gamma, beta, scale, shift);
  kF<<<256, 256, 0, stream>>>(H, scale, shift, W2, b2, Fbuf);
  kZ<<<112, 256, 0, stream>>>(Fbuf, Wfc, bfc, Z);

  // Fuse logits + variance-augmentation gather
  kOut<<<800, 256, 0, stream>>>(Z, y, ratio, D, E, out);
}